// KANLayer_60464549593380
// MI455X (gfx1250) — compile-verified
//
#include <hip/hip_runtime.h>
#include <hip/hip_bf16.h>

// Problem constants (match reference setup_inputs)
#define KAN_D      256          // input dims per row
#define KAN_N      256          // neurons
#define KAN_DEG    8            // Chebyshev degree
#define KAN_F      (KAN_D * (KAN_DEG + 1))   // 2304 features

typedef float v2f __attribute__((ext_vector_type(2)));
typedef float v8f __attribute__((ext_vector_type(8)));

#if defined(__HIP_DEVICE_COMPILE__) && defined(__has_builtin)
#if __has_builtin(__builtin_amdgcn_wmma_f32_16x16x4_f32)
#define KAN_HAVE_WMMA_F32X4 1
#endif
#endif

// -----------------------------------------------------------------------------
// Stage 1: w[f] = sum_n h[n] * coeffs[n*F + f]   (F = 2304, N = 256)
// One wave (32 threads) per 16-wide f tile -> 144 blocks of 32.
// Mapped onto V_WMMA_F32_16X16X4_F32 as D += A(16x4) * B(4x16):
//   A[m][k] = h[n0+k] * coeffs[(n0+k)*F + f0+m]   (h folded into A)
//   B       = all-ones  => every column of D equals the partial w vector,
//             making the result independent of B's exact lane layout.
// Documented 32-bit A layout: lanes 0-15 hold M=lane with K={0,1} in the two
// VGPRs; lanes 16-31 hold K={2,3}. Documented D layout: VGPR v = row M=v for
// lanes 0-15 (N=lane) and M=v+8 for lanes 16-31 (N=lane-16).
// -----------------------------------------------------------------------------
__global__ __launch_bounds__(32) void kan_stage1_reduce(
    const float* __restrict__ coeffs,   // [N, F]
    const float* __restrict__ h,        // [N]
    float* __restrict__ w)              // [F]
{
    const int lane = threadIdx.x;        // 0..31, one wave per block
    const int f0   = blockIdx.x * 16;    // 144 tiles cover F=2304
    const int m    = lane & 15;

#if defined(KAN_HAVE_WMMA_F32X4)
    const int kb = (lane >> 4) * 2;      // 0 for lanes 0-15, 2 for lanes 16-31
    v8f c = {};
    v2f bmat;
    bmat[0] = 1.0f;
    bmat[1] = 1.0f;
    for (int n0 = 0; n0 < KAN_N; n0 += 4) {
        const int r0 = n0 + kb;
        v2f a;
        a[0] = coeffs[(size_t)r0 * KAN_F + f0 + m]       * h[r0];
        a[1] = coeffs[(size_t)(r0 + 1) * KAN_F + f0 + m] * h[r0 + 1];
        // 8 args: (neg_a, A, neg_b, B, c_mod, C, reuse_a, reuse_b)
        c = __builtin_amdgcn_wmma_f32_16x16x4_f32(
                false, a, false, bmat, (short)0, c, false, false);
    }
    // Column N=0 of D carries w[f0..f0+15]: lane 0 has M=0..7, lane 16 M=8..15.
    if (lane == 0) {
#pragma unroll
        for (int v = 0; v < 8; ++v) w[f0 + v] = c[v];
    } else if (lane == 16) {
#pragma unroll
        for (int v = 0; v < 8; ++v) w[f0 + 8 + v] = c[v];
    }
#else
    // VALU fallback: lanes 0-15 each own one f column.
    if (lane < 16) {
        float acc = 0.0f;
        for (int n = 0; n < KAN_N; ++n)
            acc = fmaf(h[n], coeffs[(size_t)n * KAN_F + f0 + m], acc);
        w[f0 + m] = acc;
    }
#endif
}

// -----------------------------------------------------------------------------
// Stage 2 (memory-bound GEMV): out[b] = sum_d sum_k T_k(tanh(x[b,d])) * w[9d+k]
// One wave per row. Lane l owns d = l + 32*i (i = 0..7): b32 loads are fully
// coalesced across the wave, and the LDS reads w[9d+k] hit 32 distinct banks
// because gcd(9,64)=1. Degree-8 Chebyshev dot via Clenshaw recurrence
// (one FMA + one SUB per step), tanh via a single v_exp_f32 + v_rcp_f32.
// 8 independent Clenshaw chains per lane (unrolled) hide VALU latency.
// -----------------------------------------------------------------------------
__global__ __launch_bounds__(256) void kan_stage2_gemv(
    const float* __restrict__ x,    // [B, D]
    const float* __restrict__ w,    // [F]
    float* __restrict__ out)        // [B]
{
    __shared__ float ws[KAN_F];     // 9216 B of the 320 KB WGP LDS
    for (int i = threadIdx.x; i < KAN_F; i += 256) ws[i] = w[i];
    __syncthreads();

    const int lane = threadIdx.x & 31;
    const int wv   = threadIdx.x >> 5;          // 8 waves per block
    const int b    = (blockIdx.x << 3) + wv;    // one row per wave
    const float* xr = x + (size_t)b * KAN_D;

    float acc = 0.0f;
#pragma unroll
    for (int i = 0; i < 8; ++i) {
        const int d = lane + (i << 5);
        const float xv = xr[d];
        // tanh(x) = 1 - 2/(exp(2x)+1); saturates correctly for |x| large.
        const float e = __expf(2.0f * xv);
        const float t = fmaf(-2.0f, __builtin_amdgcn_rcpf(e + 1.0f), 1.0f);

        const float* wp = &ws[d * (KAN_DEG + 1)];
        const float t2 = t + t;
        float c1 = wp[KAN_DEG];                 // Clenshaw: b_{k}=2t*b_{k+1}-b_{k+2}+w_k
        float c2 = 0.0f;
#pragma unroll
        for (int k = KAN_DEG - 1; k >= 1; --k) {
            const float nb = fmaf(t2, c1, wp[k] - c2);
            c2 = c1;
            c1 = nb;
        }
        acc += fmaf(t, c1, wp[0] - c2);         // T0 term closes the recurrence
    }

    // wave32 reduction
#pragma unroll
    for (int off = 16; off > 0; off >>= 1)
        acc += __shfl_down(acc, off, 32);
    if (lane == 0) out[b] = acc;
}

extern "C" void kernel_launch(void* const* d_in, const int* in_sizes, int n_in,
                              void* d_out, int out_size, void* d_ws, size_t ws_size,
                              hipStream_t stream) {
    const float* x      = (const float*)d_in[0];   // [B, 256]
    const float* coeffs = (const float*)d_in[1];   // [256, 2304]
    const float* h      = (const float*)d_in[2];   // [256]
    float* out = (float*)d_out;                    // [B]
    float* w   = (float*)d_ws;                     // 2304 floats of scratch

    const int B = in_sizes[0] / KAN_D;             // 32768

    // Stage 1: fold hweights into a single feature vector (uses v_wmma f32).
    kan_stage1_reduce<<<KAN_F / 16, 32, 0, stream>>>(coeffs, h, w);
    // Stage 2: stream x once (32 MB) -> memory-bound at ~23.3 TB/s.
    kan_stage2_gemv<<<B / 8, 256, 0, stream>>>(x, w, out);
}